// LightGCNLite_47536698032634
// MI455X (gfx1250) — compile-verified
//
#include <hip/hip_runtime.h>
#include <hip/hip_bf16.h>

#define N_USERS 100000
#define N_ITEMS 50000
#define N_NODES (N_USERS + N_ITEMS)
#define NNZ     3200000
#define DIM     64
#define BATCH   8192

typedef float v2f __attribute__((ext_vector_type(2)));
typedef float v8f __attribute__((ext_vector_type(8)));

// ---------------------------------------------------------------------------
// X = concat(user_emb, item_emb0)  (float4-vectorized)
// ---------------------------------------------------------------------------
__global__ void init_x_kernel(const float* __restrict__ user_emb,
                              const float* __restrict__ item_emb0,
                              float* __restrict__ X) {
    long long i = (long long)blockIdx.x * blockDim.x + threadIdx.x;   // float4 idx
    const long long nU = (long long)N_USERS * (DIM / 4);
    const long long nT = (long long)N_NODES * (DIM / 4);
    if (i >= nT) return;
    float4 v = (i < nU) ? ((const float4*)user_emb)[i]
                        : ((const float4*)item_emb0)[i - nU];
    ((float4*)X)[i] = v;
}

// ---------------------------------------------------------------------------
// zero a float buffer (float4-vectorized)
// ---------------------------------------------------------------------------
__global__ void zero_kernel(float* __restrict__ X, long long n4) {
    long long i = (long long)blockIdx.x * blockDim.x + threadIdx.x;
    if (i >= n4) return;
    ((float4*)X)[i] = make_float4(0.f, 0.f, 0.f, 0.f);
}

// ---------------------------------------------------------------------------
// SpMM scatter: Y[dst[e]] += val[e] * X[src[e]]   (16 threads per edge,
// float4 gather per thread, 4 global f32 atomics per thread)
// ---------------------------------------------------------------------------
__global__ void spmm_kernel(const int*   __restrict__ src,
                            const int*   __restrict__ dst,
                            const float* __restrict__ val,
                            const float* __restrict__ X,
                            float*       __restrict__ Y) {
    long long t = (long long)blockIdx.x * blockDim.x + threadIdx.x;
    int e = (int)(t >> 4);
    int q = (int)(t & 15);
    if (e >= NNZ) return;
    int   s = src[e];
    int   d = dst[e];
    float v = val[e];
    float4 x = *(const float4*)(X + (long long)s * DIM + q * 4);
    float* yp = Y + (long long)d * DIM + q * 4;
    atomicAdd(yp + 0, v * x.x);
    atomicAdd(yp + 1, v * x.y);
    atomicAdd(yp + 2, v * x.z);
    atomicAdd(yp + 3, v * x.w);
}

// ---------------------------------------------------------------------------
// Sg[i] (+)= X[(N_USERS + items[i])]   — accumulate batched item rows
// ---------------------------------------------------------------------------
__global__ void gather_add_kernel(const float* __restrict__ X,
                                  const int*   __restrict__ items,
                                  float*       __restrict__ Sg,
                                  int initFlag) {
    int t = blockIdx.x * blockDim.x + threadIdx.x;     // BATCH*16 float4 slots
    if (t >= BATCH * (DIM / 4)) return;
    int i = t >> 4;
    int q = t & 15;
    long long row = (long long)(N_USERS + items[i]);
    float4 v = *(const float4*)(X + row * DIM + q * 4);
    float4* sp = (float4*)Sg + t;
    if (initFlag) {
        *sp = v;
    } else {
        float4 o = *sp;
        o.x += v.x; o.y += v.y; o.z += v.z; o.w += v.w;
        *sp = o;
    }
}

// ---------------------------------------------------------------------------
// T (8192x64) = 0.25 * Sg (8192x64) @ W^T (64x64) + b
// One wave per 16x16 tile, V_WMMA_F32_16X16X4_F32, K=64 in 16 steps.
// blockDim = 128 (4 waves): wave w -> tile_n = w; blockIdx.x -> tile_m.
// ---------------------------------------------------------------------------
__global__ void gemm_wmma_kernel(const float* __restrict__ Sg,
                                 const float* __restrict__ W,
                                 const float* __restrict__ bvec,
                                 float*       __restrict__ T) {
    const int lane   = threadIdx.x & 31;
    const int tile_n = threadIdx.x >> 5;       // 0..3  (N = 64 -> 4 tiles)
    const int tile_m = blockIdx.x;             // 0..511 (M = 8192 -> 512 tiles)

    const int mrow  = tile_m * 16 + (lane & 15);   // A row for this lane
    const int ncol  = tile_n * 16 + (lane & 15);   // B col for this lane
    const int khalf = (lane >> 4) * 2;             // lanes 0-15: K{0,1}; 16-31: K{2,3}

    v8f c = {};
    #pragma unroll
    for (int ks = 0; ks < DIM; ks += 4) {
        v2f a, bb;
        // A (16x4): Sg row-major
        a.x = Sg[(long long)mrow * DIM + ks + khalf + 0];
        a.y = Sg[(long long)mrow * DIM + ks + khalf + 1];
        // B (4x16) = W^T  =>  B[k][j] = W[j*DIM + k]
        bb.x = W[(long long)ncol * DIM + ks + khalf + 0];
        bb.y = W[(long long)ncol * DIM + ks + khalf + 1];
        c = __builtin_amdgcn_wmma_f32_16x16x4_f32(
                /*neg_a=*/false, a, /*neg_b=*/false, bb,
                /*c_mod=*/(short)0, c, /*reuse_a=*/false, /*reuse_b=*/false);
    }

    // D layout: VGPR r -> M = r + (lane>=16 ? 8 : 0), N = lane & 15
    const float bj    = bvec[ncol];
    const int   mbase = tile_m * 16 + ((lane >> 4) ? 8 : 0);
    #pragma unroll
    for (int r = 0; r < 8; ++r) {
        T[(long long)(mbase + r) * DIM + ncol] = 0.25f * c[r] + bj;
    }
}

// ---------------------------------------------------------------------------
// gamma[i] = user_emb[users[i]] . T[i]
// ---------------------------------------------------------------------------
__global__ void dot_kernel(const float* __restrict__ user_emb,
                           const int*   __restrict__ users,
                           const float* __restrict__ T,
                           float*       __restrict__ out) {
    int i = blockIdx.x * blockDim.x + threadIdx.x;
    if (i >= BATCH) return;
    const float4* u = (const float4*)(user_emb + (long long)users[i] * DIM);
    const float4* t = (const float4*)(T + (long long)i * DIM);
    float s = 0.f;
    #pragma unroll
    for (int q = 0; q < DIM / 4; ++q) {
        float4 a = u[q], b4 = t[q];
        s += a.x * b4.x + a.y * b4.y + a.z * b4.z + a.w * b4.w;
    }
    out[i] = s;
}

// ---------------------------------------------------------------------------
extern "C" void kernel_launch(void* const* d_in, const int* in_sizes, int n_in,
                              void* d_out, int out_size, void* d_ws, size_t ws_size,
                              hipStream_t stream) {
    const float* user_emb  = (const float*)d_in[0];   // 100000 x 64
    const float* item_emb0 = (const float*)d_in[1];   // 50000 x 64
    const float* edge_val  = (const float*)d_in[2];   // NNZ
    const float* W         = (const float*)d_in[3];   // 64 x 64
    const float* bvec      = (const float*)d_in[4];   // 64
    const int*   edge_src  = (const int*)d_in[5];     // NNZ
    const int*   edge_dst  = (const int*)d_in[6];     // NNZ
    const int*   users     = (const int*)d_in[7];     // BATCH
    const int*   items     = (const int*)d_in[8];     // BATCH
    float*       out       = (float*)d_out;           // BATCH

    // workspace layout
    float* Xa = (float*)d_ws;                          // N_NODES*64
    float* Xb = Xa + (size_t)N_NODES * DIM;            // N_NODES*64
    float* Sg = Xb + (size_t)N_NODES * DIM;            // BATCH*64 (sum X1+X2+X3 items)
    float* T  = Sg + (size_t)BATCH * DIM;              // BATCH*64

    const long long n4    = (long long)N_NODES * (DIM / 4);       // float4 count
    const int       gridX = (int)((n4 + 255) / 256);

    init_x_kernel<<<gridX, 256, 0, stream>>>(user_emb, item_emb0, Xa);

    float* Xcur = Xa;
    float* Xnxt = Xb;
    for (int layer = 0; layer < 3; ++layer) {
        zero_kernel<<<gridX, 256, 0, stream>>>(Xnxt, n4);

        long long spmm_threads = (long long)NNZ * 16;
        int spmm_grid = (int)((spmm_threads + 255) / 256);
        spmm_kernel<<<spmm_grid, 256, 0, stream>>>(edge_src, edge_dst, edge_val,
                                                   Xcur, Xnxt);

        int ga_grid = (BATCH * (DIM / 4) + 255) / 256;
        gather_add_kernel<<<ga_grid, 256, 0, stream>>>(Xnxt, items, Sg,
                                                       layer == 0 ? 1 : 0);
        float* tmp = Xcur; Xcur = Xnxt; Xnxt = tmp;
    }

    // T = 0.25 * Sg @ W^T + b   (WMMA f32)
    gemm_wmma_kernel<<<BATCH / 16, 128, 0, stream>>>(Sg, W, bvec, T);

    // gamma = rowwise dot(user_emb[users], T)
    dot_kernel<<<BATCH / 256, 256, 0, stream>>>(user_emb, users, T, out);
}